// GuidanceDistanceLoss_25924422599412
// MI455X (gfx1250) — compile-verified
//
#include <hip/hip_runtime.h>
#include <math.h>

#define NB   2
#define DD   96
#define HH   192
#define WW   192
#define DHW  (DD * HH * WW)      // 3,538,944 per field
#define KT   25                   // gaussian taps (sigma=3, truncated=4 -> tail=12)
#define KRAD 12
#define RB   1024                 // stage-1 reduction blocks per field

typedef __attribute__((ext_vector_type(2))) float v2f;
typedef __attribute__((ext_vector_type(8))) float v8f;

// MONAI gaussian_1d(sigma=3, approx='erf'): integrate gaussian over unit bins.
__device__ __forceinline__ float gw25(int t) {
    const float tc = 0.70710678f / 3.0f;   // 1/(sigma*sqrt(2))
    float x = (float)(t - KRAD);
    float k = 0.5f * (erff(tc * (x + 0.5f)) - erff(tc * (x - 0.5f)));
    return k < 0.f ? 0.f : k;
}

// One-time: 25 Gaussian taps into workspace (keeps erf code out of hot kernel).
__global__ void taps_kernel(float* __restrict__ taps) {
    int t = threadIdx.x;
    if (t < KT) taps[t] = gw25(t);
}

// Pass 1: S = rint(o0)+rint(o1)  (RoundSTE fwd, channel-collapsed),
//         G = (g0!=0)+(g1!=0)    (channel-collapsed guidance mask)
__global__ void prep_kernel(const float* __restrict__ outs,
                            const float* __restrict__ guid,
                            float* __restrict__ S, float* __restrict__ G) {
    int i = blockIdx.x * blockDim.x + threadIdx.x;
    if (i >= NB * DHW) return;
    int f = i / DHW;
    int r = i - f * DHW;
    int base = f * 2 * DHW + r;                 // [B,C,D,H,W], C=2
    S[i] = rintf(outs[base]) + rintf(outs[base + DHW]);
    float g0 = guid[base], g1 = guid[base + DHW];
    G[i] = (g0 != 0.f ? 1.f : 0.f) + (g1 != 0.f ? 1.f : 0.f);
}

// Pass 2: Sobel magnitude (zero-padded cross-correlation, per reference kernels)
__global__ void sobel_kernel(const float* __restrict__ S, float* __restrict__ E) {
    int i = blockIdx.x * blockDim.x + threadIdx.x;
    if (i >= NB * DHW) return;
    int f = i / DHW;
    int r = i - f * DHW;
    int d = r / (HH * WW);
    int rem = r - d * HH * WW;
    int h = rem / WW;
    int w = rem - h * WW;
    const float* Sf = S + f * DHW;
    auto ld = [&](int dd, int hh, int ww) -> float {
        if ((unsigned)dd >= (unsigned)DD || (unsigned)hh >= (unsigned)HH ||
            (unsigned)ww >= (unsigned)WW) return 0.f;
        return Sf[(dd * HH + hh) * WW + ww];
    };
    float p00 = ld(d, h - 1, w - 1), p01 = ld(d, h - 1, w), p02 = ld(d, h - 1, w + 1);
    float p10 = ld(d, h,     w - 1),                        p12 = ld(d, h,     w + 1);
    float p20 = ld(d, h + 1, w - 1), p21 = ld(d, h + 1, w), p22 = ld(d, h + 1, w + 1);
    float q00 = ld(d - 1, h - 1, w), q02 = ld(d - 1, h + 1, w);
    float q20 = ld(d + 1, h - 1, w), q22 = ld(d + 1, h + 1, w);
    // kv = gradH (-1,0,1) x smoothW (1,2,1)
    float xv = (p20 + 2.f * p21 + p22) - (p00 + 2.f * p01 + p02);
    // kh = smoothH (1,2,1) x gradW (-1,0,1)
    float xh = (p02 + 2.f * p12 + p22) - (p00 + 2.f * p10 + p20);
    // kd = smoothD (1,2,1) x gradH (-1,0,1)
    float xd = (q02 - q00) + 2.f * (p21 - p01) + (q22 - q20);
    E[i] = sqrtf(xv * xv + xh * xh + xd * xd + 1e-6f);
}

// 25-tap 1-D Gaussian conv along one axis as implicit banded GEMM on the WMMA
// pipe: per 16x16 output tile, D = A[16x40] * B[40x16] via ten
// V_WMMA_F32_16X16X4_F32. The A window is staged into LDS with
// GLOBAL_LOAD_ASYNC_TO_LDS_B32 (ASYNCcnt path, no VGPR round-trip); zero
// padding cells are plain DS stores to disjoint addresses. Lane mapping and
// store path are chosen so the stride-1 axis is always lane-contiguous.
__global__ __launch_bounds__(32) void gauss_conv_wmma(
    const float* __restrict__ in, float* __restrict__ out,
    const float* __restrict__ taps,
    int conv_len, int conv_stride, int row_stride, int outer_stride,
    int outer_cnt) {
    __shared__ float tile[16 * 41];   // A window; stride 41 coprime 64 banks
    __shared__ float outt[16 * 17];   // store-coalescing tile for row_stride==1
    __shared__ float kco[KT];
    int tid   = threadIdx.x;
    int f     = blockIdx.z / outer_cnt;
    int outer = blockIdx.z - f * outer_cnt;
    int base  = f * DHW + outer * outer_stride;
    int c0    = blockIdx.x * 16;      // conv-axis output tile origin
    int r0    = blockIdx.y * 16;      // row-axis tile origin
    int cs1   = (conv_stride == 1);   // uniform across grid

    if (tid < KT) kco[tid] = taps[tid];

    // Stage 16 rows x 40 window columns (conv positions c0-12 .. c0+27).
    // cs1: lanes sweep j (conv axis, stride 1)  -> contiguous global runs.
    // else: lanes sweep r (row axis, stride 1)  -> contiguous global runs.
    for (int idx = tid; idx < 16 * 40; idx += 32) {
        int r, j;
        if (cs1) { r = idx / 40; j = idx - r * 40; }
        else     { j = idx >> 4; r = idx & 15; }
        int cp = c0 + j - KRAD;
        unsigned lo = (unsigned)(size_t)&tile[r * 41 + j];
        if (cp >= 0 && cp < conv_len) {
            const float* gp = in + base + cp * conv_stride + (r0 + r) * row_stride;
            asm volatile("global_load_async_to_lds_b32 %0, %1, off"
                         :: "v"(lo), "v"((unsigned long long)(size_t)gp)
                         : "memory");
        } else {
            tile[r * 41 + j] = 0.f;   // zero padding (disjoint from async dests)
        }
    }
    asm volatile("s_wait_asynccnt 0x0" ::: "memory");
    __syncthreads();

    // f32 16x16x4 WMMA lane layout: lanes 0-15 -> K pair (4k,4k+1),
    // lanes 16-31 -> K pair (4k+2,4k+3); M (A) / N (B) = lane & 15.
    int m  = tid & 15;
    int hi = (tid >> 4) << 1;          // 0 or 2
    v8f acc = {};
#pragma unroll
    for (int kc = 0; kc < 10; ++kc) {
        int jb = 4 * kc + hi;
        v2f a, b;
        a.x = tile[m * 41 + jb];
        a.y = tile[m * 41 + jb + 1];
        int t0 = jb - m;               // banded: B[j,n] = k[j-n], 0<=j-n<25
        int t1 = t0 + 1;
        b.x = (t0 >= 0 && t0 < KT) ? kco[t0] : 0.f;
        b.y = (t1 >= 0 && t1 < KT) ? kco[t1] : 0.f;
        acc = __builtin_amdgcn_wmma_f32_16x16x4_f32(
            false, a, false, b, (short)0, acc, false, false);
    }

    // D layout: VGPR v -> M = v (lanes 0-15) / v+8 (lanes 16-31), N = lane&15
    int mb = (tid >> 4) * 8;
    if (cs1) {
        // lanes sweep N (conv axis, stride 1) -> contiguous 16-float stores
#pragma unroll
        for (int v = 0; v < 8; ++v)
            out[base + (c0 + m) * conv_stride + (r0 + v + mb) * row_stride] = acc[v];
    } else {
        // stage through LDS so stores sweep the stride-1 row axis
#pragma unroll
        for (int v = 0; v < 8; ++v)
            outt[m * 17 + v + mb] = acc[v];          // outt[n][M]
        __syncthreads();
        for (int idx = tid; idx < 256; idx += 32) {
            int r = idx & 15, n = idx >> 4;
            out[base + (c0 + n) * conv_stride + (r0 + r)] = outt[n * 17 + r];
        }
    }
}

// Deterministic two-stage reduction: sum(Xs*Gt) and sum(Gt) per field.
__global__ void reduce1_kernel(const float* __restrict__ X,
                               const float* __restrict__ G,
                               float* __restrict__ part) {
    __shared__ float s0[256], s1[256];
    int f = blockIdx.y;
    const float* Xf = X + f * DHW;
    const float* Gf = G + f * DHW;
    float a0 = 0.f, a1 = 0.f;
    for (int i = blockIdx.x * 256 + threadIdx.x; i < DHW; i += RB * 256) {
        float g = Gf[i];
        a0 += Xf[i] * g;
        a1 += g;
    }
    s0[threadIdx.x] = a0; s1[threadIdx.x] = a1;
    __syncthreads();
    for (int st = 128; st > 0; st >>= 1) {
        if (threadIdx.x < st) {
            s0[threadIdx.x] += s0[threadIdx.x + st];
            s1[threadIdx.x] += s1[threadIdx.x + st];
        }
        __syncthreads();
    }
    if (threadIdx.x == 0) {
        part[(f * RB + blockIdx.x) * 2 + 0] = s0[0];
        part[(f * RB + blockIdx.x) * 2 + 1] = s1[0];
    }
}

__global__ void reduce2_kernel(const float* __restrict__ part,
                               float* __restrict__ outp) {
    __shared__ float s0[256], s1[256];
    int f = blockIdx.x;
    float a0 = 0.f, a1 = 0.f;
    for (int j = threadIdx.x; j < RB; j += 256) {
        a0 += part[(f * RB + j) * 2 + 0];
        a1 += part[(f * RB + j) * 2 + 1];
    }
    s0[threadIdx.x] = a0; s1[threadIdx.x] = a1;
    __syncthreads();
    for (int st = 128; st > 0; st >>= 1) {
        if (threadIdx.x < st) {
            s0[threadIdx.x] += s0[threadIdx.x + st];
            s1[threadIdx.x] += s1[threadIdx.x + st];
        }
        __syncthreads();
    }
    if (threadIdx.x == 0) outp[f] = expf(-s0[0] / (s1[0] + 1.0f));
}

extern "C" void kernel_launch(void* const* d_in, const int* in_sizes, int n_in,
                              void* d_out, int out_size, void* d_ws, size_t ws_size,
                              hipStream_t stream) {
    (void)in_sizes; (void)n_in; (void)out_size; (void)ws_size;
    const float* outs = (const float*)d_in[0];   // [2,2,96,192,192] f32
    const float* guid = (const float*)d_in[1];   // [2,2,96,192,192] f32
    float* out  = (float*)d_out;                 // [2] f32
    float* buf0 = (float*)d_ws;                  // S -> smooth tmp -> Xs
    float* buf1 = buf0 + (size_t)NB * DHW;       // gsum -> smooth tmp
    float* buf2 = buf1 + (size_t)NB * DHW;       // E -> smooth tmp -> Gt
    float* part = buf2 + (size_t)NB * DHW;       // 2*RB*2 floats of partials
    float* taps = part + (size_t)NB * RB * 2;    // 25 floats of gaussian taps

    int n = NB * DHW;
    taps_kernel <<<1, 32, 0, stream>>>(taps);
    prep_kernel <<<(n + 255) / 256, 256, 0, stream>>>(outs, guid, buf0, buf1);
    sobel_kernel<<<(n + 255) / 256, 256, 0, stream>>>(buf0, buf2);

    // grid: (conv tiles, row tiles, outer * NB)
    dim3 gW(WW / 16, HH / 16, DD * NB);   // conv along W, rows along H, outer d
    dim3 gH(HH / 16, WW / 16, DD * NB);   // conv along H, rows along W, outer d
    dim3 gD(DD / 16, WW / 16, HH * NB);   // conv along D, rows along W, outer h

    // Smooth edge field: buf2 -> buf0 -> buf2 -> buf0 (= Xs)
    gauss_conv_wmma<<<gW, 32, 0, stream>>>(buf2, buf0, taps, WW, 1,       WW, HH * WW, DD);
    gauss_conv_wmma<<<gH, 32, 0, stream>>>(buf0, buf2, taps, HH, WW,      1,  HH * WW, DD);
    gauss_conv_wmma<<<gD, 32, 0, stream>>>(buf2, buf0, taps, DD, HH * WW, 1,  WW,      HH);
    // Smooth guidance field: buf1 -> buf2 -> buf1 -> buf2 (= Gt)
    gauss_conv_wmma<<<gW, 32, 0, stream>>>(buf1, buf2, taps, WW, 1,       WW, HH * WW, DD);
    gauss_conv_wmma<<<gH, 32, 0, stream>>>(buf2, buf1, taps, HH, WW,      1,  HH * WW, DD);
    gauss_conv_wmma<<<gD, 32, 0, stream>>>(buf1, buf2, taps, DD, HH * WW, 1,  WW,      HH);

    dim3 gr(RB, NB);
    reduce1_kernel<<<gr, 256, 0, stream>>>(buf0, buf2, part);
    reduce2_kernel<<<NB, 256, 0, stream>>>(part, out);
}